// MinkowskiFlow_45681272160506
// MI455X (gfx1250) — compile-verified
//
#include <hip/hip_runtime.h>

typedef float v2f __attribute__((ext_vector_type(2)));
typedef float v8f __attribute__((ext_vector_type(8)));

#define DDIM 64
#define TILE 16
#define WAVES_PER_BLOCK 4
#define ROWS_PER_BLOCK (TILE * WAVES_PER_BLOCK) // 64
#define MCHUNKS 4
#define LDSW 68 // padded LDS row stride (words) -> conflict-free b64 reads

// ---------------------------------------------------------------------------
// Kernel 1: L2-normalize feature rows (wave per row, 32 lanes x 2 floats)
// ---------------------------------------------------------------------------
__global__ void mf_normalize(const float* __restrict__ fs,
                             const float* __restrict__ ft,
                             float* __restrict__ fsN,
                             float* __restrict__ ftN,
                             int rowsS, int rowsT) {
  int wave = (int)((blockIdx.x * blockDim.x + threadIdx.x) >> 5);
  int lane = threadIdx.x & 31;
  if (wave >= rowsS + rowsT) return;
  const float* src;
  float* dst;
  if (wave < rowsS) {
    src = fs + (size_t)wave * DDIM;
    dst = fsN + (size_t)wave * DDIM;
  } else {
    src = ft + (size_t)(wave - rowsS) * DDIM;
    dst = ftN + (size_t)(wave - rowsS) * DDIM;
  }
  v2f v = *(const v2f*)(src + lane * 2);
  float ss = v.x * v.x + v.y * v.y;
#pragma unroll
  for (int m = 16; m >= 1; m >>= 1) ss += __shfl_xor(ss, m, 32);
  float rinv = rsqrtf(ss);
  v2f o;
  o.x = v.x * rinv;
  o.y = v.y * rinv;
  *(v2f*)(dst + lane * 2) = o;
}

// ---------------------------------------------------------------------------
// Kernel 2: fused kernel-correlation.  Each wave: one 16-row N tile.
// Block (4 waves) stages ft tiles through LDS.  FP32 WMMA (16x16x4) chain.
// ---------------------------------------------------------------------------
__global__ void __launch_bounds__(WAVES_PER_BLOCK * 32)
mf_main(const float* __restrict__ fsN, const float* __restrict__ ftN,
        const float* __restrict__ coorS, const float* __restrict__ coorT,
        const float* __restrict__ epsP, float* __restrict__ partial,
        int B, int N, int M) {
  __shared__ float sB[TILE * LDSW];
  __shared__ float sC[TILE * 4];

  const int tid = threadIdx.x;
  const int lane = tid & 31;
  const int wave = tid >> 5;
  const int half = lane >> 4; // 0: K even pair, 1: K odd pair (per ISA A/B layout)
  const int l16 = lane & 15;

  const int nBlocks = N / ROWS_PER_BLOCK;
  const int bid = blockIdx.x;
  const int chunk = bid % MCHUNKS;
  const int nb = (bid / MCHUNKS) % nBlocks;
  const int b = bid / (MCHUNKS * nBlocks);

  const int nBase = nb * ROWS_PER_BLOCK + wave * TILE;
  const int mLen = M / MCHUNKS;
  const int mStart = chunk * mLen;

  const float tau = __expf(epsP[0]) + 0.03f;
  const float scale = 2.0f / tau; // K = exp(scale * (fdot - 1))

  // --- A operand (16 rows x 64 K, fp32): lane<16 holds K=4k,4k+1; lane>=16 K=4k+2,4k+3
  const int aRow = nBase + l16;
  const float* aPtr = fsN + ((size_t)(b * N + aRow)) * DDIM + (half << 1);
  v2f aF[16];
#pragma unroll
  for (int kk = 0; kk < 16; ++kk) aF[kk] = *(const v2f*)(aPtr + kk * 4);

  // --- coordinate A operand (K = x,y,z,0)
  const float* csPtr = coorS + ((size_t)(b * N + aRow)) * 3;
  v2f aC;
  aC.x = csPtr[half ? 2 : 0];
  aC.y = half ? 0.0f : csPtr[1];

  // --- |coor_s|^2 for the 8 output rows this lane accumulates (C-layout rows)
  float cs2[8];
#pragma unroll
  for (int r = 0; r < 8; ++r) {
    const float* p = coorS + ((size_t)(b * N + nBase + r + half * 8)) * 3;
    cs2[r] = p[0] * p[0] + p[1] * p[1] + p[2] * p[2];
  }

  float rs[8], ax[8], ay[8], az[8];
#pragma unroll
  for (int r = 0; r < 8; ++r) { rs[r] = 0.f; ax[r] = 0.f; ay[r] = 0.f; az[r] = 0.f; }

  for (int m0 = 0; m0 < mLen; m0 += TILE) {
    const int mBase = mStart + m0;
    __syncthreads();
    // --- cooperative stage: 16x64 ft tile (padded rows) + 16 target coords
    {
      int row = tid >> 3, seg = tid & 7;
      const float4* src =
          (const float4*)(ftN + ((size_t)(b * M + mBase + row)) * DDIM + seg * 8);
      float4* dst = (float4*)(sB + row * LDSW + seg * 8);
      dst[0] = src[0];
      dst[1] = src[1];
      if (tid < TILE) {
        const float* ct = coorT + ((size_t)(b * M + mBase + tid)) * 3;
        sC[tid * 4 + 0] = ct[0];
        sC[tid * 4 + 1] = ct[1];
        sC[tid * 4 + 2] = ct[2];
        sC[tid * 4 + 3] = 0.f;
      }
    }
    __syncthreads();

    // --- feature dot: chain 16 fp32 WMMAs over K=64
    const float* bPtr = sB + l16 * LDSW + (half << 1);
    v8f fdot = {};
#pragma unroll
    for (int kk = 0; kk < 16; ++kk) {
      v2f bF = *(const v2f*)(bPtr + kk * 4);
      fdot = __builtin_amdgcn_wmma_f32_16x16x4_f32(false, aF[kk], false, bF,
                                                   (short)0, fdot, false, false);
    }
    // --- coordinate dot (K = x,y,z,0)
    v2f bC;
    bC.x = sC[l16 * 4 + (half ? 2 : 0)];
    bC.y = half ? 0.0f : sC[l16 * 4 + 1];
    v8f cdot = {};
    cdot = __builtin_amdgcn_wmma_f32_16x16x4_f32(false, aC, false, bC, (short)0,
                                                 cdot, false, false);

    const float ctx = sC[l16 * 4 + 0];
    const float cty = sC[l16 * 4 + 1];
    const float ctz = sC[l16 * 4 + 2];
    const float ct2 = ctx * ctx + cty * cty + ctz * ctz;

#pragma unroll
    for (int r = 0; r < 8; ++r) {
      float d2 = cs2[r] + ct2 - 2.0f * cdot[r];
      float k = (d2 < 100.0f) ? __expf(scale * (fdot[r] - 1.0f)) : 0.0f;
      rs[r] += k;
      ax[r] += k * ctx;
      ay[r] += k * cty;
      az[r] += k * ctz;
    }
  }

  // --- reduce across the 16 columns held by each wave half
#pragma unroll
  for (int r = 0; r < 8; ++r) {
#pragma unroll
    for (int m = 8; m >= 1; m >>= 1) {
      rs[r] += __shfl_xor(rs[r], m, 32);
      ax[r] += __shfl_xor(ax[r], m, 32);
      ay[r] += __shfl_xor(ay[r], m, 32);
      az[r] += __shfl_xor(az[r], m, 32);
    }
  }
  if (l16 == 0) {
#pragma unroll
    for (int r = 0; r < 8; ++r) {
      int row = nBase + r + half * 8;
      float4* outp = (float4*)partial + ((size_t)(b * N + row)) * MCHUNKS + chunk;
      *outp = make_float4(rs[r], ax[r], ay[r], az[r]);
    }
  }
}

// ---------------------------------------------------------------------------
// Kernel 3: combine M-chunk partials, divide, subtract coor_s
// ---------------------------------------------------------------------------
__global__ void mf_finalize(const float* __restrict__ partial,
                            const float* __restrict__ coorS,
                            float* __restrict__ out, int total) {
  int i = blockIdx.x * blockDim.x + threadIdx.x;
  if (i >= total) return;
  float rs = 0.f, ax = 0.f, ay = 0.f, az = 0.f;
  const float4* p = (const float4*)partial + (size_t)i * MCHUNKS;
#pragma unroll
  for (int c = 0; c < MCHUNKS; ++c) {
    float4 v = p[c];
    rs += v.x;
    ax += v.y;
    ay += v.z;
    az += v.w;
  }
  float inv = 1.0f / (rs + 1e-8f);
  out[i * 3 + 0] = ax * inv - coorS[i * 3 + 0];
  out[i * 3 + 1] = ay * inv - coorS[i * 3 + 1];
  out[i * 3 + 2] = az * inv - coorS[i * 3 + 2];
}

extern "C" void kernel_launch(void* const* d_in, const int* in_sizes, int n_in,
                              void* d_out, int out_size, void* d_ws, size_t ws_size,
                              hipStream_t stream) {
  const float* feat_s = (const float*)d_in[0];
  const float* feat_t = (const float*)d_in[1];
  const float* coor_s = (const float*)d_in[2];
  const float* coor_t = (const float*)d_in[3];
  const float* epsP = (const float*)d_in[4];
  float* out = (float*)d_out;

  const int B = 2;
  const int N = in_sizes[0] / (B * DDIM);
  const int M = in_sizes[1] / (B * DDIM);

  float* ws = (float*)d_ws;
  float* fsN = ws;
  float* ftN = fsN + (size_t)B * N * DDIM;
  float* partial = ftN + (size_t)B * M * DDIM; // B*N*MCHUNKS float4s

  // normalize: one wave per row, 8 waves (256 threads) per block
  const int rowsS = B * N, rowsT = B * M;
  const int totalRows = rowsS + rowsT;
  mf_normalize<<<(totalRows + 7) / 8, 256, 0, stream>>>(feat_s, feat_t, fsN, ftN,
                                                        rowsS, rowsT);

  const int grid = B * (N / ROWS_PER_BLOCK) * MCHUNKS;
  mf_main<<<grid, WAVES_PER_BLOCK * 32, 0, stream>>>(fsN, ftN, coor_s, coor_t,
                                                     epsP, partial, B, N, M);

  const int total = B * N;
  mf_finalize<<<(total + 255) / 256, 256, 0, stream>>>(partial, coor_s, out, total);
}